// DigitCaps_4398046511414
// MI455X (gfx1250) — compile-verified
//
#include <hip/hip_runtime.h>

typedef __attribute__((ext_vector_type(2))) float v2f;
typedef __attribute__((ext_vector_type(8))) float v8f;

#define R_DIM 1024
#define C_DIM 10
#define O_DIM 16
#define I_DIM 8
#define B_DIM 256
#define CO    160            // C_DIM * O_DIM
#define RCHUNKS 8
#define RCHUNK  128          // R_DIM / RCHUNKS

// ---------------- zero init (b_ij must be 0 at start of every call) ----------
__global__ void zero_kernel(float* __restrict__ p, int n) {
    int t = blockIdx.x * blockDim.x + threadIdx.x;
    if (t < n) p[t] = 0.0f;
}

// ---------------- transpose x[B,I,R] -> xt[R,I,B] ---------------------------
__global__ void xpose_kernel(const float* __restrict__ x, float* __restrict__ xt) {
    int t = blockIdx.x * blockDim.x + threadIdx.x;   // linear index into xt
    if (t >= R_DIM * I_DIM * B_DIM) return;
    int b = t & (B_DIM - 1);
    int i = (t >> 8) & (I_DIM - 1);
    int r = t >> 11;
    xt[t] = x[((size_t)b * I_DIM + i) * R_DIM + r];
}

// ---------------- u_hat GEMM via V_WMMA_F32_16X16X4_F32 ---------------------
// Per route r: U[r] (160 x 256) = W[r] (160 x 8) @ xt[r] (8 x 256)
// U layout: [r][co][b]  (b contiguous -> coalesced for both routing reductions)
__global__ void __launch_bounds__(256) uhat_wmma_kernel(const float* __restrict__ W,
                                                        const float* __restrict__ xt,
                                                        float* __restrict__ U) {
    __shared__ float Asl[CO * I_DIM];      // 160 x 8  (W[r], co-major)
    __shared__ float Bsl[I_DIM * B_DIM];   // 8 x 256  (xt[r])

    const int r   = blockIdx.x;
    const int tid = threadIdx.x;

    const float* Wr = W  + (size_t)r * (CO * I_DIM);
    const float* Xr = xt + (size_t)r * (I_DIM * B_DIM);
    for (int k = tid; k < CO * I_DIM;    k += 256) Asl[k] = Wr[k];
    for (int k = tid; k < I_DIM * B_DIM; k += 256) Bsl[k] = Xr[k];
    __syncthreads();

    const int wave = tid >> 5;
    const int lane = tid & 31;
    const int half = lane >> 4;        // 0: lanes 0-15, 1: lanes 16-31
    const int ln   = lane & 15;
    const int kb   = half * 2;         // K base for this half (A: K, B: K rows)

    float* Ur = U + (size_t)r * CO * B_DIM;

    // 10 M-tiles x 16 N-tiles = 160 tiles over 8 waves
    for (int tile = wave; tile < 160; tile += 8) {
        const int mt = tile >> 4;      // 0..9
        const int nt = tile & 15;      // 0..15

        // A operand: 16x4 fp32 in 2 VGPRs. lane<16: (M=ln,K=kb..kb+1); lane>=16: K=2..3
        const int arow = mt * 16 + ln;
        v2f a0, a1, b0, b1;
        a0.x = Asl[arow * I_DIM + kb + 0];
        a0.y = Asl[arow * I_DIM + kb + 1];
        a1.x = Asl[arow * I_DIM + kb + 4];
        a1.y = Asl[arow * I_DIM + kb + 5];

        // B operand: 4x16 fp32 in 2 VGPRs, rows half-split across lanes like C/D
        const int bcol = nt * 16 + ln;
        b0.x = Bsl[(kb + 0) * B_DIM + bcol];
        b0.y = Bsl[(kb + 1) * B_DIM + bcol];
        b1.x = Bsl[(kb + 4) * B_DIM + bcol];
        b1.y = Bsl[(kb + 5) * B_DIM + bcol];

        v8f acc = {};
        acc = __builtin_amdgcn_wmma_f32_16x16x4_f32(false, a0, false, b0,
                                                    (short)0, acc, false, false);
        acc = __builtin_amdgcn_wmma_f32_16x16x4_f32(false, a1, false, b1,
                                                    (short)0, acc, false, false);

        // D: VGPR j holds (M = j + half*8, N = ln)
        #pragma unroll
        for (int j = 0; j < 8; ++j) {
            int m = mt * 16 + j + half * 8;
            Ur[(size_t)m * B_DIM + nt * 16 + ln] = acc[j];
        }
    }
}

// ---------------- softmax over C (per route) --------------------------------
__global__ void softmax_kernel(const float* __restrict__ bij, float* __restrict__ cij) {
    int r = blockIdx.x * blockDim.x + threadIdx.x;
    if (r >= R_DIM) return;
    float row[C_DIM];
    float mx = -3.402823466e38f;
    #pragma unroll
    for (int c = 0; c < C_DIM; ++c) { row[c] = bij[r * C_DIM + c]; mx = fmaxf(mx, row[c]); }
    float s = 0.0f;
    #pragma unroll
    for (int c = 0; c < C_DIM; ++c) { row[c] = __expf(row[c] - mx); s += row[c]; }
    float inv = 1.0f / s;
    #pragma unroll
    for (int c = 0; c < C_DIM; ++c) cij[r * C_DIM + c] = row[c] * inv;
}

// ---------------- s_j partial reduction over an R-chunk ---------------------
// part[chunk][co][b] = sum over 128 routes of c_ij[r,c] * U[r][co][b]
__global__ void __launch_bounds__(256) sj_partial_kernel(const float* __restrict__ U,
                                                         const float* __restrict__ cij,
                                                         float* __restrict__ part) {
    const int co    = blockIdx.x;      // 0..159
    const int chunk = blockIdx.y;      // 0..7
    const int c     = co >> 4;
    const int b     = threadIdx.x;

    const float* Ub = U + (size_t)co * B_DIM + b;
    float acc = 0.0f;
    const int r0 = chunk * RCHUNK;
    #pragma unroll 4
    for (int rr = 0; rr < RCHUNK; ++rr) {
        int r = r0 + rr;
        acc = fmaf(cij[r * C_DIM + c], Ub[(size_t)r * (CO * B_DIM)], acc);
    }
    part[((size_t)chunk * CO + co) * B_DIM + b] = acc;
}

// ---------------- combine partials + squash over O --------------------------
// v layout: [co][b]
__global__ void __launch_bounds__(256) squash_kernel(const float* __restrict__ part,
                                                     float* __restrict__ v) {
    const int c = blockIdx.x;          // 0..9
    const int b = threadIdx.x;
    float s[O_DIM];
    float sn = 0.0f;
    #pragma unroll
    for (int o = 0; o < O_DIM; ++o) {
        float acc = 0.0f;
        #pragma unroll
        for (int k = 0; k < RCHUNKS; ++k)
            acc += part[((size_t)k * CO + c * O_DIM + o) * B_DIM + b];
        s[o] = acc;
        sn += acc * acc;
    }
    float scale = sn / ((1.0f + sn) * sqrtf(sn));
    #pragma unroll
    for (int o = 0; o < O_DIM; ++o)
        v[(size_t)(c * O_DIM + o) * B_DIM + b] = s[o] * scale;
}

// ---------------- agreement: b_ij[r,c] += mean_b <u_hat, v> -----------------
__global__ void __launch_bounds__(256) agree_kernel(const float* __restrict__ U,
                                                    const float* __restrict__ v,
                                                    float* __restrict__ bij) {
    const int r   = blockIdx.x;
    const int c   = blockIdx.y;
    const int tid = threadIdx.x;

    const float* Ur = U + ((size_t)r * CO + c * O_DIM) * B_DIM;
    const float* vc = v + (size_t)c * O_DIM * B_DIM;

    float acc = 0.0f;
    #pragma unroll 4
    for (int k = tid; k < O_DIM * B_DIM; k += 256)
        acc = fmaf(Ur[k], vc[k], acc);

    // wave32 shuffle reduction
    #pragma unroll
    for (int off = 16; off > 0; off >>= 1)
        acc += __shfl_down(acc, off, 32);

    __shared__ float red[8];
    if ((tid & 31) == 0) red[tid >> 5] = acc;
    __syncthreads();
    if (tid == 0) {
        float t = 0.0f;
        #pragma unroll
        for (int w = 0; w < 8; ++w) t += red[w];
        bij[r * C_DIM + c] += t * (1.0f / (float)B_DIM);
    }
}

// ---------------- final output: out[b,c,o,1] = v[co][b] ---------------------
__global__ void out_kernel(const float* __restrict__ v, float* __restrict__ out) {
    int t = blockIdx.x * blockDim.x + threadIdx.x;
    if (t >= B_DIM * CO) return;
    int b  = t / CO;
    int co = t % CO;
    out[t] = v[(size_t)co * B_DIM + b];
}

extern "C" void kernel_launch(void* const* d_in, const int* in_sizes, int n_in,
                              void* d_out, int out_size, void* d_ws, size_t ws_size,
                              hipStream_t stream) {
    const float* x = (const float*)d_in[0];   // [256, 8, 1024]
    const float* W = (const float*)d_in[1];   // [1024, 10, 16, 8]
    float* out = (float*)d_out;               // [256, 10, 16, 1] = 40960 floats

    // workspace carve-up (floats)
    float* ws   = (float*)d_ws;
    float* xt   = ws;                                   //  2,097,152  (xt[R][I][B])
    float* U    = xt   + (size_t)R_DIM * I_DIM * B_DIM; // 41,943,040  (u_hat[R][CO][B])
    float* bij  = U    + (size_t)R_DIM * CO * B_DIM;    //     10,240
    float* cij  = bij  + R_DIM * C_DIM;                 //     10,240
    float* part = cij  + R_DIM * C_DIM;                 //    327,680
    float* v    = part + (size_t)RCHUNKS * CO * B_DIM;  //     40,960

    zero_kernel<<<(R_DIM * C_DIM + 255) / 256, 256, 0, stream>>>(bij, R_DIM * C_DIM);

    xpose_kernel<<<(R_DIM * I_DIM * B_DIM + 255) / 256, 256, 0, stream>>>(x, xt);

    uhat_wmma_kernel<<<R_DIM, 256, 0, stream>>>(W, xt, U);

    for (int it = 0; it < 3; ++it) {
        softmax_kernel<<<(R_DIM + 255) / 256, 256, 0, stream>>>(bij, cij);
        sj_partial_kernel<<<dim3(CO, RCHUNKS), 256, 0, stream>>>(U, cij, part);
        squash_kernel<<<C_DIM, 256, 0, stream>>>(part, v);
        if (it < 2)
            agree_kernel<<<dim3(R_DIM, C_DIM), 256, 0, stream>>>(U, v, bij);
    }

    out_kernel<<<(B_DIM * CO + 255) / 256, 256, 0, stream>>>(v, out);
}